// GraphConvLayer_7335804142016
// MI455X (gfx1250) — compile-verified
//
#include <hip/hip_runtime.h>

#define DIM 128
#define NPAD 50048  // padded node count for workspace layout (64B aligned)

typedef float v2f __attribute__((ext_vector_type(2)));
typedef float v8f __attribute__((ext_vector_type(8)));

// ---------------------------------------------------------------- degree ----
__global__ __launch_bounds__(256) void zero_u32_kernel(unsigned* __restrict__ p, int n) {
    int i = blockIdx.x * 256 + threadIdx.x;
    if (i < n) p[i] = 0u;
}

__global__ __launch_bounds__(256) void deg_kernel(const int* __restrict__ row,
                                                  unsigned* __restrict__ deg, int E) {
    int i = blockIdx.x * 256 + threadIdx.x;
    if (i < E) atomicAdd(&deg[row[i]], 1u);
}

__global__ __launch_bounds__(256) void dis_kernel(const unsigned* __restrict__ deg,
                                                  float* __restrict__ dis, int n) {
    int i = blockIdx.x * 256 + threadIdx.x;
    if (i < n) dis[i] = rsqrtf((float)deg[i] + 1.0f);  // +1 = self loop; always > 0
}

// ------------------------------------------------- fp32 WMMA GEMM + epilogue ----
// Block: 256 threads = 8 waves. Block b -> rows [b*16, b*16+16); wave w -> cols [w*16, w*16+16).
// Computes t2 = dis[m] * (x @ W)  and  out = bias + dis[m]^2 * (x @ W)  (self-loop init).
__global__ __launch_bounds__(256) void gemm_wmma_kernel(const float* __restrict__ x,
                                                        const float* __restrict__ w,
                                                        const float* __restrict__ bias,
                                                        const float* __restrict__ dis,
                                                        float* __restrict__ t2,
                                                        float* __restrict__ out) {
    __shared__ float lx[16][132];  // 132-float row stride: bank-conflict-free, 16B aligned

    const int m0 = blockIdx.x * 16;
    const int t  = threadIdx.x;

    // Cooperative load of the 16x128 A tile (each thread: 8 contiguous floats).
    {
        const int lr = t >> 4;          // 0..15
        const int lc = (t & 15) << 3;   // 0,8,...,120
        const float* src = x + (size_t)(m0 + lr) * DIM + lc;
        float4 p0 = *(const float4*)(src);
        float4 p1 = *(const float4*)(src + 4);
        *(float4*)(&lx[lr][lc])     = p0;
        *(float4*)(&lx[lr][lc + 4]) = p1;
    }
    __syncthreads();

    const int wv   = t >> 5;        // wave id 0..7 -> column tile
    const int lane = t & 31;
    const int idx  = lane & 15;     // M (for A) / N (for B,C,D)
    const int half = lane >> 4;     // selects K pair
    const int n0   = wv * 16;

    v8f c = {0.f, 0.f, 0.f, 0.f, 0.f, 0.f, 0.f, 0.f};

    #pragma unroll 8
    for (int kb = 0; kb < DIM; kb += 4) {
        const int k0 = kb + 2 * half;
        // A: lane holds x[m0+idx][k0], x[m0+idx][k0+1]   (ISA 16x4 fp32 A layout)
        v2f a = *(const v2f*)(&lx[idx][k0]);
        // B: lane holds W[k0][n0+idx], W[k0+1][n0+idx]   (4x16 fp32 B layout)
        v2f b;
        b.x = w[(size_t)k0 * DIM + n0 + idx];
        b.y = w[(size_t)(k0 + 1) * DIM + n0 + idx];
        // (neg_a, A, neg_b, B, c_mod, C, reuse_a, reuse_b)
        c = __builtin_amdgcn_wmma_f32_16x16x4_f32(false, a, false, b, (short)0, c,
                                                  false, false);
    }

    // Epilogue: C VGPR r, lane L -> element (M = r + 8*half, N = idx).
    const int n   = n0 + idx;
    const float bn = bias[n];
    #pragma unroll
    for (int r = 0; r < 8; ++r) {
        const int m   = m0 + r + 8 * half;
        const float dm = dis[m];
        const float v  = c[r];
        const float sv = dm * v;                     // dis[m] * transformed
        t2 [(size_t)m * DIM + n] = sv;
        out[(size_t)m * DIM + n] = bn + dm * sv;     // bias + dis[m]^2 * transformed
    }
}

// ---------------------------------------------------------------- scatter ----
// One wave per edge: out[col] += dis[col] * t2[row]  (t2 already carries dis[row]).
__global__ __launch_bounds__(256) void scatter_kernel(const int* __restrict__ row,
                                                      const int* __restrict__ col,
                                                      const float* __restrict__ dis,
                                                      const float* __restrict__ t2,
                                                      float* __restrict__ out, int E) {
    const int gid  = blockIdx.x * 256 + threadIdx.x;
    const int e    = gid >> 5;
    const int lane = gid & 31;
    if (e >= E) return;

    const int r = row[e];
    const int c = col[e];
    const float dc = dis[c];

    const float4 v = *(const float4*)(t2 + (size_t)r * DIM + lane * 4);
    float* op = out + (size_t)c * DIM + lane * 4;
    unsafeAtomicAdd(op + 0, dc * v.x);
    unsafeAtomicAdd(op + 1, dc * v.y);
    unsafeAtomicAdd(op + 2, dc * v.z);
    unsafeAtomicAdd(op + 3, dc * v.w);
}

// ----------------------------------------------------------------- launch ----
extern "C" void kernel_launch(void* const* d_in, const int* in_sizes, int n_in,
                              void* d_out, int out_size, void* d_ws, size_t ws_size,
                              hipStream_t stream) {
    const float* x    = (const float*)d_in[0];
    const int*   ei   = (const int*)d_in[1];
    const float* w    = (const float*)d_in[2];
    const float* bias = (const float*)d_in[3];
    float*       out  = (float*)d_out;

    const int nodes = in_sizes[0] / DIM;     // 50000 (divisible by 16)
    const int E     = in_sizes[1] / 2;       // 800000
    const int* row  = ei;                    // edge_index[0]
    const int* col  = ei + E;                // edge_index[1]

    unsigned* deg = (unsigned*)d_ws;
    float*    dis = (float*)d_ws + NPAD;
    float*    t2  = (float*)d_ws + 2 * NPAD;

    const int nb_nodes = (nodes + 255) / 256;
    zero_u32_kernel<<<nb_nodes, 256, 0, stream>>>(deg, nodes);
    deg_kernel<<<(E + 255) / 256, 256, 0, stream>>>(row, deg, E);
    dis_kernel<<<nb_nodes, 256, 0, stream>>>(deg, dis, nodes);
    gemm_wmma_kernel<<<nodes / 16, 256, 0, stream>>>(x, w, bias, dis, t2, out);
    scatter_kernel<<<(E * 32 + 255) / 256, 256, 0, stream>>>(row, col, dis, t2, out, E);
}